// APM_p_Group_45938970198646
// MI455X (gfx1250) — compile-verified
//
#include <hip/hip_runtime.h>
#include <math.h>

typedef float v2f __attribute__((ext_vector_type(2)));
typedef float v4f __attribute__((ext_vector_type(4)));
typedef float v8f __attribute__((ext_vector_type(8)));

#define NPTS 16384
#define KNN 12

__device__ __forceinline__ bool knn_better(float d1, int i1, float d2, int i2) {
  return (d1 < d2) || (d1 == d2 && i1 < i2);
}

// Filter one 16-candidate tile (8 rows per lane) against the top-12 threshold,
// doing the rare sorted-insert only when the fmin-tree says something qualifies.
__device__ __forceinline__ void knn_process_tile(const v8f& D, int c, int cofs, float qsq,
                                                 float (&kd)[KNN], int (&ki)[KNN],
                                                 float& thr) {
  float m0 = fminf(D[0], D[1]), m1 = fminf(D[2], D[3]);
  float m2 = fminf(D[4], D[5]), m3 = fminf(D[6], D[7]);
  float dmin = fminf(fminf(m0, m1), fminf(m2, m3));
  if (dmin < thr) {
#pragma unroll
    for (int v = 0; v < 8; ++v) {
      float d2 = D[v] + qsq;
      if (d2 < kd[KNN - 1]) {
        float cd = d2; int cx = c + cofs + v;   // fully unrolled sorted insert
#pragma unroll
        for (int j = 0; j < KNN; ++j) {
          bool bt = cd < kd[j];
          float od = kd[j]; int oi = ki[j];
          if (bt) { kd[j] = cd; ki[j] = cx; cd = od; cx = oi; }
        }
      }
    }
    thr = kd[KNN - 1] - qsq;
  }
}

// ---------------------------------------------------------------------------
// Kernel 1: brute-force KNN via V_WMMA_F32_16X16X4_F32 + feature build.
// Point cloud lives in LDS as float4 (x, y, z, |p|^2) = 256KB of 320KB WGP LDS.
// A (candidates, 16x4) = (x, y, z, |p|^2); B (queries, 4x16) = (-2x,-2y,-2z, 1)
//   => D[m][q] = |p_c|^2 - 2 p_c.p_q ;  d2 = D + |p_q|^2 (per-lane scalar).
// D layout: lane L<16 holds rows 0..7 (col L), lane L+16 holds rows 8..15.
// Candidate loop unrolled x2, A-operands prefetched one iteration ahead.
// Each lane keeps a sorted top-12 in registers; halves merged in LDS.
// ---------------------------------------------------------------------------
__global__ __launch_bounds__(256) void knn_feat_kernel(const float* __restrict__ p,
                                                       float* __restrict__ X) {
  extern __shared__ float lds[];
  v4f*   pts    = (v4f*)lds;                      // 16384 x float4
  float* mergeD = lds + 4 * NPTS;                 // 8 waves * 16 q * 24 entries
  int*   mergeI = (int*)(mergeD + 8 * 16 * 24);

  const int tid = threadIdx.x;
  for (int i = tid; i < NPTS; i += 256) {
    float x = p[i * 3 + 0], y = p[i * 3 + 1], z = p[i * 3 + 2];
    v4f q; q.x = x; q.y = y; q.z = z; q.w = x * x + y * y + z * z;
    pts[i] = q;
  }
  __syncthreads();

  const int wave   = tid >> 5;
  const int lane   = tid & 31;
  const int qlocal = lane & 15;
  const int hi     = lane >> 4;               // 0: lanes 0-15, 1: lanes 16-31
  const int cofs   = hi * 8;
  const int gq     = blockIdx.x * 128 + wave * 16 + qlocal;

  const v4f  qp  = pts[gq];
  const float qx = qp.x, qy = qp.y, qz = qp.z, qsq = qp.w;

  // B matrix (queries) is loop-invariant.
  v2f B;
  if (hi == 0) { B.x = -2.0f * qx; B.y = -2.0f * qy; }
  else         { B.x = -2.0f * qz; B.y = 1.0f; }

  float kd[KNN]; int ki[KNN];
#pragma unroll
  for (int j = 0; j < KNN; ++j) { kd[j] = 3.0e38f; ki[j] = 0x7fffffff; }
  float thr = 1.0e38f;                        // kd[11] - qsq, in raw-D space

  // lane's A-operand slice: .xy for lanes 0-15, .zw for lanes 16-31
  const char* pbase = (const char*)pts + qlocal * 16 + hi * 8;

  v2f A0 = *(const v2f*)(pbase);              // prefetch tiles 0 and 1
  v2f A1 = *(const v2f*)(pbase + 256);
  for (int c = 0; c < NPTS; c += 32) {
    const int c2 = (c + 32) & (NPTS - 1);     // wrapped: always legal addresses
    const int c3 = (c + 48) & (NPTS - 1);
    v2f N0 = *(const v2f*)(pbase + c2 * 16);  // prefetch next pair of tiles
    v2f N1 = *(const v2f*)(pbase + c3 * 16);

    v8f Cz = {0.f, 0.f, 0.f, 0.f, 0.f, 0.f, 0.f, 0.f};
    v8f D0 = __builtin_amdgcn_wmma_f32_16x16x4_f32(false, A0, false, B,
                                                   (short)0, Cz, false, false);
    knn_process_tile(D0, c, cofs, qsq, kd, ki, thr);

    v8f D1 = __builtin_amdgcn_wmma_f32_16x16x4_f32(false, A1, false, B,
                                                   (short)0, Cz, false, false);
    knn_process_tile(D1, c + 16, cofs, qsq, kd, ki, thr);

    A0 = N0; A1 = N1;
  }

  // dump both half-lists, then lanes 0-15 merge their query's two sorted lists
  const int base = (wave * 16 + qlocal) * 24 + hi * 12;
#pragma unroll
  for (int j = 0; j < KNN; ++j) { mergeD[base + j] = kd[j]; mergeI[base + j] = ki[j]; }
  __syncthreads();

  if (hi == 0) {
    const int b0 = (wave * 16 + qlocal) * 24;
    int i1 = 0, i2 = 0;
    int outi[KNN];
#pragma unroll
    for (int j = 0; j < KNN; ++j) {
      float d1 = mergeD[b0 + i1];      int x1 = mergeI[b0 + i1];
      float d2 = mergeD[b0 + 12 + i2]; int x2 = mergeI[b0 + 12 + i2];
      bool t = knn_better(d1, x1, d2, x2);
      outi[j] = t ? x1 : x2;
      if (t) ++i1; else ++i2;
    }
    // features: [p, |p - n1|, ..., |p - n11|] (entry 0 == self, dropped)
    float* Xr = X + gq * 36;
    Xr[0] = qx; Xr[1] = qy; Xr[2] = qz;
#pragma unroll
    for (int j = 1; j < KNN; ++j) {
      v4f nb = pts[outi[j]];
      Xr[3 * j + 0] = fabsf(qx - nb.x);
      Xr[3 * j + 1] = fabsf(qy - nb.y);
      Xr[3 * j + 2] = fabsf(qz - nb.z);
    }
  }
}

// ---------------------------------------------------------------------------
// Fused (bn(prev)+relu) -> conv -> partial BN stats (LDS atomics -> global).
// a == nullptr means no input normalization (layer 1).
// ---------------------------------------------------------------------------
template <int CIN, int COUT>
__global__ __launch_bounds__(256) void conv_bn_stats_kernel(
    const float* __restrict__ in, const float* __restrict__ W,
    const float* __restrict__ a, const float* __restrict__ bsh,
    float* __restrict__ out, float* __restrict__ S, float* __restrict__ Q) {
  __shared__ float sw[COUT * CIN];
  __shared__ float ssum[COUT], ssq[COUT];
  const int tid = threadIdx.x;
  for (int i = tid; i < COUT * CIN; i += 256) sw[i] = W[i];
  if (tid < COUT) { ssum[tid] = 0.0f; ssq[tid] = 0.0f; }
  __syncthreads();

  const int n = blockIdx.x * 256 + tid;
  float xv[CIN];
#pragma unroll
  for (int i = 0; i < CIN; ++i) {
    float v = in[n * CIN + i];
    if (a) v = fmaxf(a[i] * v + bsh[i], 0.0f);
    xv[i] = v;
  }
#pragma unroll
  for (int c = 0; c < COUT; ++c) {
    float y = 0.0f;
#pragma unroll
    for (int i = 0; i < CIN; ++i) y += sw[c * CIN + i] * xv[i];
    out[n * COUT + c] = y;
    atomicAdd(&ssum[c], y);
    atomicAdd(&ssq[c], y * y);
  }
  __syncthreads();
  if (tid < COUT) { atomicAdd(&S[tid], ssum[tid]); atomicAdd(&Q[tid], ssq[tid]); }
}

template <int C>
__global__ void bn_finalize_kernel(const float* __restrict__ S, const float* __restrict__ Q,
                                   const float* __restrict__ g, const float* __restrict__ b,
                                   float* __restrict__ pa, float* __restrict__ pb) {
  int c = threadIdx.x;
  if (c < C) {
    const float invN = 1.0f / 16384.0f;
    float mu  = S[c] * invN;
    float var = Q[c] * invN - mu * mu;
    float aa  = g[c] * rsqrtf(var + 1e-5f);
    pa[c] = aa;
    pb[c] = b[c] - aa * mu;
  }
}

// bn3 + relu + linear head + softmax over the B=4 axis (one thread per column)
__global__ __launch_bounds__(256) void head_softmax_kernel(
    const float* __restrict__ Y3, const float* __restrict__ a3, const float* __restrict__ b3,
    const float* __restrict__ wr, const float* __restrict__ br, float* __restrict__ outp) {
  const int m = blockIdx.x * 256 + threadIdx.x;  // 0..4095
  float r[4];
#pragma unroll
  for (int b = 0; b < 4; ++b) {
    int row = b * 4096 + m;
    float h0 = fmaxf(a3[0] * Y3[row * 3 + 0] + b3[0], 0.0f);
    float h1 = fmaxf(a3[1] * Y3[row * 3 + 1] + b3[1], 0.0f);
    float h2 = fmaxf(a3[2] * Y3[row * 3 + 2] + b3[2], 0.0f);
    r[b] = h0 * wr[0] + h1 * wr[1] + h2 * wr[2] + br[0];
  }
  float mx = fmaxf(fmaxf(r[0], r[1]), fmaxf(r[2], r[3]));
  float e0 = expf(r[0] - mx), e1 = expf(r[1] - mx);
  float e2 = expf(r[2] - mx), e3 = expf(r[3] - mx);
  float inv = 1.0f / (e0 + e1 + e2 + e3);
  outp[0 * 4096 + m] = e0 * inv;
  outp[1 * 4096 + m] = e1 * inv;
  outp[2 * 4096 + m] = e2 * inv;
  outp[3 * 4096 + m] = e3 * inv;
}

__global__ void zero_kernel(float* __restrict__ p, int n) {
  int i = blockIdx.x * blockDim.x + threadIdx.x;
  if (i < n) p[i] = 0.0f;
}

extern "C" void kernel_launch(void* const* d_in, const int* in_sizes, int n_in,
                              void* d_out, int out_size, void* d_ws, size_t ws_size,
                              hipStream_t stream) {
  const float* p  = (const float*)d_in[0];
  const float* w1 = (const float*)d_in[1];
  const float* g1 = (const float*)d_in[2];
  const float* b1 = (const float*)d_in[3];
  const float* w2 = (const float*)d_in[4];
  const float* g2 = (const float*)d_in[5];
  const float* b2 = (const float*)d_in[6];
  const float* w3 = (const float*)d_in[7];
  const float* g3 = (const float*)d_in[8];
  const float* b3 = (const float*)d_in[9];
  const float* wr = (const float*)d_in[10];
  const float* br = (const float*)d_in[11];
  float* out = (float*)d_out;

  float* ws  = (float*)d_ws;
  float* X   = ws;                    // 16384*36
  float* Y1  = X  + NPTS * 36;        // 16384*18
  float* Y2  = Y1 + NPTS * 18;        // 16384*9
  float* Y3  = Y2 + NPTS * 9;         // 16384*3
  float* S1  = Y3 + NPTS * 3;  float* Q1 = S1 + 18;
  float* S2  = Q1 + 18;        float* Q2 = S2 + 9;
  float* S3  = Q2 + 9;         float* Q3 = S3 + 3;
  float* a1  = Q3 + 3;         float* bb1 = a1 + 18;
  float* a2  = bb1 + 18;       float* bb2 = a2 + 9;
  float* a3  = bb2 + 9;        float* bb3 = a3 + 3;

  zero_kernel<<<1, 64, 0, stream>>>(S1, 60);                 // S1..Q3 contiguous

  const size_t ldsz = (size_t)(4 * NPTS + 8 * 16 * 24 * 2) * 4;  // 286720 B
  knn_feat_kernel<<<128, 256, ldsz, stream>>>(p, X);

  conv_bn_stats_kernel<36, 18><<<64, 256, 0, stream>>>(X,  w1, nullptr, nullptr, Y1, S1, Q1);
  bn_finalize_kernel<18><<<1, 32, 0, stream>>>(S1, Q1, g1, b1, a1, bb1);
  conv_bn_stats_kernel<18, 9><<<64, 256, 0, stream>>>(Y1, w2, a1, bb1, Y2, S2, Q2);
  bn_finalize_kernel<9><<<1, 32, 0, stream>>>(S2, Q2, g2, b2, a2, bb2);
  conv_bn_stats_kernel<9, 3><<<64, 256, 0, stream>>>(Y2, w3, a2, bb2, Y3, S3, Q3);
  bn_finalize_kernel<3><<<1, 32, 0, stream>>>(S3, Q3, g3, b3, a3, bb3);

  head_softmax_kernel<<<16, 256, 0, stream>>>(Y3, a3, bb3, wr, br, out);
}